// MoEGate_77395310674356
// MI455X (gfx1250) — compile-verified
//
#include <hip/hip_runtime.h>
#include <hip/hip_bf16.h>

typedef float v2f __attribute__((ext_vector_type(2)));
typedef float v8f __attribute__((ext_vector_type(8)));

#define H       2048
#define E       64
#define TOPK    8
#define BSZ     4
#define SEQ     4096
#define NTOK    (BSZ * SEQ)
#define TPW     32                    // tokens per wave
#define WPB     4                     // waves per block
#define TPB     (TPW * WPB)           // tokens per block (128)
#define LROW    65                    // padded LDS row stride (floats) -> conflict-free

// ---------------------------------------------------------------------------
// Zero the aux-loss accumulators in workspace (graph-replay deterministic).
// ---------------------------------------------------------------------------
__global__ void moe_gate_init(float* ws) {
    int i = threadIdx.x;              // 512 threads: 2 * BSZ * E = 512 floats
    ws[i] = 0.0f;
}

// ---------------------------------------------------------------------------
// Main: fp32 WMMA GEMM (logits) + softmax + top-8 + aux-loss partials.
// grid = NTOK/TPB blocks of 128 threads (4 waves); wave handles 32 tokens.
// ---------------------------------------------------------------------------
__global__ __launch_bounds__(TPB) void moe_gate_main(
        const float* __restrict__ x,      // [NTOK, H]
        const float* __restrict__ w,      // [E, H]
        float* __restrict__ out_idx,      // [NTOK, 8] (indices stored as float)
        float* __restrict__ out_wt,       // [NTOK, 8]
        float* __restrict__ g_ssum,       // [BSZ, E] softmax score sums
        float* __restrict__ g_cnt)        // [BSZ, E] top-k counts
{
    __shared__ float lgt[WPB][TPW * LROW];   // per-wave logits/exp staging
    __shared__ float invs[WPB][TPW];         // per-token 1/sum(exp)
    __shared__ float blk_ssum[E];
    __shared__ float blk_cnt[E];

    const int tid  = threadIdx.x;
    const int wave = tid >> 5;
    const int lane = tid & 31;

    if (tid < E) { blk_ssum[tid] = 0.0f; blk_cnt[tid] = 0.0f; }

    const int tok0 = blockIdx.x * TPB + wave * TPW;

    // ---------------- GEMM: logits[32 tok][64 exp], K = 2048 ----------------
    // Fragment layout (wave32, 16x16x4 f32): lanes 0-15 hold K={0,1} in
    // {v0,v1}; lanes 16-31 hold K={2,3}. Same pattern for A (lane=M) and B
    // (lane=N), so one float2 load at row*H + k + 2*(lane>>4) per fragment.
    const int m     = lane & 15;
    const int khalf = (lane >> 4) << 1;

    const float* a0 = x + (size_t)(tok0 +  0 + m) * H + khalf;
    const float* a1 = x + (size_t)(tok0 + 16 + m) * H + khalf;
    const float* b0 = w + (size_t)( 0 + m) * H + khalf;
    const float* b1 = w + (size_t)(16 + m) * H + khalf;
    const float* b2 = w + (size_t)(32 + m) * H + khalf;
    const float* b3 = w + (size_t)(48 + m) * H + khalf;

    v8f acc[2][4];
    #pragma unroll
    for (int i = 0; i < 2; ++i)
        #pragma unroll
        for (int j = 0; j < 4; ++j)
            acc[i][j] = (v8f){0.f, 0.f, 0.f, 0.f, 0.f, 0.f, 0.f, 0.f};

    #pragma unroll 4
    for (int k = 0; k < H; k += 4) {
        v2f fa0 = *(const v2f*)(a0 + k);
        v2f fa1 = *(const v2f*)(a1 + k);
        v2f fb0 = *(const v2f*)(b0 + k);
        v2f fb1 = *(const v2f*)(b1 + k);
        v2f fb2 = *(const v2f*)(b2 + k);
        v2f fb3 = *(const v2f*)(b3 + k);

        acc[0][0] = __builtin_amdgcn_wmma_f32_16x16x4_f32(false, fa0, false, fb0, (short)0, acc[0][0], false, false);
        acc[0][1] = __builtin_amdgcn_wmma_f32_16x16x4_f32(false, fa0, false, fb1, (short)0, acc[0][1], false, false);
        acc[0][2] = __builtin_amdgcn_wmma_f32_16x16x4_f32(false, fa0, false, fb2, (short)0, acc[0][2], false, false);
        acc[0][3] = __builtin_amdgcn_wmma_f32_16x16x4_f32(false, fa0, false, fb3, (short)0, acc[0][3], false, false);
        acc[1][0] = __builtin_amdgcn_wmma_f32_16x16x4_f32(false, fa1, false, fb0, (short)0, acc[1][0], false, false);
        acc[1][1] = __builtin_amdgcn_wmma_f32_16x16x4_f32(false, fa1, false, fb1, (short)0, acc[1][1], false, false);
        acc[1][2] = __builtin_amdgcn_wmma_f32_16x16x4_f32(false, fa1, false, fb2, (short)0, acc[1][2], false, false);
        acc[1][3] = __builtin_amdgcn_wmma_f32_16x16x4_f32(false, fa1, false, fb3, (short)0, acc[1][3], false, false);
    }

    // C/D layout: VGPR j -> M=j (lanes 0-15) / M=j+8 (lanes 16-31), N=lane&15.
    #pragma unroll
    for (int mt = 0; mt < 2; ++mt)
        #pragma unroll
        for (int nt = 0; nt < 4; ++nt)
            #pragma unroll
            for (int j = 0; j < 8; ++j) {
                int tl = mt * 16 + j + ((lane >> 4) << 3);
                int e  = (lane & 15) + nt * 16;
                lgt[wave][tl * LROW + e] = acc[mt][nt][j];
            }
    __syncthreads();

    // ---------------- softmax (one lane per token) ----------------
    float* row = &lgt[wave][lane * LROW];
    float mx = row[0];
    #pragma unroll
    for (int e = 1; e < E; ++e) mx = fmaxf(mx, row[e]);
    float s = 0.0f;
    #pragma unroll
    for (int e = 0; e < E; ++e) {
        float v = __expf(row[e] - mx);
        row[e] = v;
        s += v;
    }
    const float inv = 1.0f / s;       // exp sums are strictly positive
    invs[wave][lane] = inv;
    __syncthreads();

    // ---------------- per-expert score sums (before top-k clobbers LDS) -----
    {
        float s0 = 0.0f, s1 = 0.0f;
        #pragma unroll
        for (int t = 0; t < TPW; ++t) {
            float iv = invs[wave][t];
            s0 += lgt[wave][t * LROW + lane]      * iv;   // bank = (t+lane)%64
            s1 += lgt[wave][t * LROW + lane + 32] * iv;
        }
        atomicAdd(&blk_ssum[lane],      s0);
        atomicAdd(&blk_ssum[lane + 32], s1);
    }
    __syncthreads();

    // ---------------- top-8 (one lane per token, destroys row) --------------
    {
        const int n = tok0 + lane;
        float topv[TOPK];
        int   topi[TOPK];
        float sum8 = 0.0f;
        #pragma unroll
        for (int k = 0; k < TOPK; ++k) {
            float bv = -1.0f;
            int   bi = 0;
            #pragma unroll
            for (int e = 0; e < E; ++e) {
                float v = row[e];
                if (v > bv) { bv = v; bi = e; }   // first index wins ties
            }
            row[bi] = -1.0f;
            float sc = bv * inv;
            topv[k] = sc;
            topi[k] = bi;
            sum8 += sc;
        }
        const float r = 1.0f / (sum8 + 1e-20f);
        #pragma unroll
        for (int k = 0; k < TOPK; ++k) {
            out_idx[(size_t)n * TOPK + k] = (float)topi[k];
            out_wt [(size_t)n * TOPK + k] = topv[k] * r;
        }
        #pragma unroll
        for (int k = 0; k < TOPK; ++k)
            atomicAdd(&blk_cnt[topi[k]], 1.0f);
    }
    __syncthreads();

    // ---------------- flush block partials (block lies in one batch) --------
    if (tid < E) {
        const int b = (blockIdx.x * TPB) / SEQ;
        atomicAdd(&g_ssum[b * E + tid], blk_ssum[tid]);
        atomicAdd(&g_cnt [b * E + tid], blk_cnt[tid]);
    }
}

// ---------------------------------------------------------------------------
// Finalize: aux = 0.1 * mean_b( sum_e ce_norm[b][e] * scores_seq_mean[b][e] )
// ---------------------------------------------------------------------------
__global__ void moe_gate_finalize(const float* __restrict__ g_ssum,
                                  const float* __restrict__ g_cnt,
                                  float* __restrict__ out_aux)
{
    __shared__ float red[E];
    const int e = threadIdx.x;        // 64 threads
    float p = 0.0f;
    #pragma unroll
    for (int b = 0; b < BSZ; ++b) {
        float ce = g_cnt [b * E + e] * ((float)E / (float)(SEQ * TOPK));
        float sm = g_ssum[b * E + e] * (1.0f / (float)SEQ);
        p += ce * sm;
    }
    red[e] = p;
    __syncthreads();
    if (e == 0) {
        float t = 0.0f;
        for (int i = 0; i < E; ++i) t += red[i];
        *out_aux = t * (0.1f / (float)BSZ);
    }
}

// ---------------------------------------------------------------------------
extern "C" void kernel_launch(void* const* d_in, const int* in_sizes, int n_in,
                              void* d_out, int out_size, void* d_ws, size_t ws_size,
                              hipStream_t stream) {
    const float* x = (const float*)d_in[0];   // [4,4096,2048] f32
    const float* w = (const float*)d_in[1];   // [64,2048]     f32

    float* out     = (float*)d_out;
    float* out_idx = out;                                 // NTOK*8
    float* out_wt  = out + (size_t)NTOK * TOPK;           // NTOK*8
    float* out_aux = out + (size_t)2 * NTOK * TOPK;       // 1

    float* g_ssum = (float*)d_ws;             // [BSZ*E]
    float* g_cnt  = g_ssum + BSZ * E;         // [BSZ*E]

    moe_gate_init<<<1, 2 * BSZ * E, 0, stream>>>(g_ssum);
    moe_gate_main<<<NTOK / TPB, TPB, 0, stream>>>(x, w, out_idx, out_wt,
                                                  g_ssum, g_cnt);
    moe_gate_finalize<<<1, E, 0, stream>>>(g_ssum, g_cnt, out_aux);
}